// CA_global_non_WFCA_40415642255793
// MI455X (gfx1250) — compile-verified
//
#include <hip/hip_runtime.h>

// CDNA5 / gfx1250 fused windowed channel-attention kernel.
// x: [8, 64, 256, 256] f32, w_qkv: [192, 64] f32, out: [8, 64, 256, 256] f32.
// One wave32 processes one 2x8 pixel tile (= 4 windows of 2x2 = 16 tokens):
//   Phase 0: async global -> LDS DMA of the x tile [64ch x 16px]  (ASYNCcnt)
//   Phase 1: QKV projection with v_wmma_f32_16x16x4_f32 (M=16 tok, N=16 ch, K=64)
//   Phase 2: per (window, head): S^T = K x Q^T (1 WMMA), in-lane softmax,
//            O = P x V (4 chained K=4 WMMAs, P^T->A via lane permute)
//   Phase 3: async LDS -> global DMA of the output tile
// Memory-bound problem (268 MB mandatory traffic @ 23.3 TB/s ~ 11.5 us), so we
// stay in f32 and fuse everything; qkv never touches HBM. The async LDS path
// removes the VGPR staging in phases 0/3 and cuts register pressure.

typedef float v2f __attribute__((ext_vector_type(2)));
typedef float v8f __attribute__((ext_vector_type(8)));

#define WAVES_PER_WG 2

__global__ __launch_bounds__(WAVES_PER_WG * 32)
void ca_fused_wmma(const float* __restrict__ x,
                   const float* __restrict__ w,
                   float* __restrict__ out) {
  // Per-wave LDS region: 1024 floats x-tile/out-tile (aliased) + 3072 floats qkv.
  __shared__ float lds[WAVES_PER_WG * 4096];

  const int lane = threadIdx.x & 31;
  const int wv   = threadIdx.x >> 5;
  const int tile = blockIdx.x * WAVES_PER_WG + wv;   // 32768 tiles total
  const int b    = tile >> 12;                       // 4096 tiles per image
  const int rem  = tile & 4095;
  const int ty   = rem >> 5;                         // 0..127 (window row)
  const int tx   = rem & 31;                         // 0..31  (4-window col grp)
  const int y0   = ty * 2;
  const int x0   = tx * 8;

  float* xbuf = lds + wv * 4096;   // [ch 0..63][pix 0..15]
  float* qkv  = xbuf + 1024;       // [o 0..191][tok 0..15]
  float* obuf = xbuf;              // reuse x region for output tile

  const int half = lane >> 4;      // which half of the wave
  const int m    = lane & 15;      // row index (A) / column index (B)
  // token m -> pixel offset inside tile: win=m>>2, hsp=(m>>1)&1, wsp=m&1
  const int po   = ((m >> 1) & 1) * 8 + (m >> 2) * 2 + (m & 1);

  const size_t imgBase = (size_t)b * 64 * 65536;

  // warm W into the cache hierarchy (global_prefetch_b8)
  __builtin_prefetch(w + (threadIdx.x << 5), 0, 0);

  // ---- Phase 0: async DMA x tile (64 ch x 2 rows x 8 px) into LDS ----
#pragma unroll
  for (int it = 0; it < 8; ++it) {
    const int idx = it * 32 + lane;
    const int ch = idx >> 2, rr = (idx >> 1) & 1, qq = idx & 1;
    const float* gp = x + imgBase + (size_t)ch * 65536
                      + (y0 + rr) * 256 + x0 + qq * 4;
    // low 32 bits of a generic pointer to LDS == addrspace(3) offset
    const unsigned ldst = (unsigned)(uintptr_t)(xbuf + ch * 16 + rr * 8 + qq * 4);
    asm volatile("global_load_async_to_lds_b128 %0, %1, off"
                 :: "v"(ldst), "v"(gp) : "memory");
  }
  asm volatile("s_wait_asynccnt 0x0" ::: "memory");

  // ---- Phase 1: QKV = X(16x64) @ W^T(64x192), 12 chunks x 16 K-steps ----
  for (int chunk = 0; chunk < 12; ++chunk) {
    v8f acc = {0.f, 0.f, 0.f, 0.f, 0.f, 0.f, 0.f, 0.f};
    const int o = chunk * 16 + m;          // output channel for this lane (B col)
#pragma unroll
    for (int s = 0; s < 16; ++s) {
      const int c0 = 4 * s + 2 * half;     // K rows: lanes<16 -> k0,k1; >=16 -> k2,k3
      v2f a;
      a.x = xbuf[c0 * 16 + po];            // A = X: row = token m, cols = channels
      a.y = xbuf[(c0 + 1) * 16 + po];
      const float2 wv2 = *(const float2*)(w + o * 64 + c0);  // W[o][c0], W[o][c0+1]
      v2f bf; bf.x = wv2.x; bf.y = wv2.y;
      acc = __builtin_amdgcn_wmma_f32_16x16x4_f32(
          false, a, false, bf, (short)0, acc, false, false);
    }
    // D layout: VGPR r, lanes<16: tok=r; lanes>=16: tok=8+r. Stage as [o][tok].
#pragma unroll
    for (int r = 0; r < 8; ++r)
      qkv[o * 16 + r + 8 * half] = acc[r];
  }
  asm volatile("s_wait_dscnt 0x0" ::: "memory");

  // ---- Phase 2: attention per (window, head) ----
  for (int win = 0; win < 4; ++win) {
    const int t0 = win * 4;
    for (int hd = 0; hd < 4; ++hd) {
      // S^T = K(16x4) x Q^T(4x16) * SCALE ; channel ch = c*4 + hd in qkv rows
      const int okk = (64 + m * 4 + hd) * 16 + t0 + 2 * half;  // K, A frag
      const int oqq = (m * 4 + hd) * 16 + t0 + 2 * half;       // Q, B frag
      const float2 kf = *(const float2*)(qkv + okk);           // ds_load_b64
      const float2 qf = *(const float2*)(qkv + oqq);
      v2f ka; ka.x = kf.x;         ka.y = kf.y;
      v2f qb; qb.x = qf.x * 0.25f; qb.y = qf.y * 0.25f;        // head_dim^-0.5
      v8f st = {0.f, 0.f, 0.f, 0.f, 0.f, 0.f, 0.f, 0.f};
      st = __builtin_amdgcn_wmma_f32_16x16x4_f32(
          false, ka, false, qb, (short)0, st, false, false);

      // softmax over ck = rows of S^T: 8 regs in-lane + xor-16 exchange
      float mx = st[0];
#pragma unroll
      for (int r = 1; r < 8; ++r) mx = fmaxf(mx, st[r]);
      mx = fmaxf(mx, __shfl_xor(mx, 16, 32));
      float sum = 0.f;
#pragma unroll
      for (int r = 0; r < 8; ++r) { st[r] = __expf(st[r] - mx); sum += st[r]; }
      sum += __shfl_xor(sum, 16, 32);
      const float inv = __builtin_amdgcn_rcpf(sum);
#pragma unroll
      for (int r = 0; r < 8; ++r) st[r] *= inv;   // st = P^T[ck][cq]

      // O(16x4) = P(16x16) @ V(16x4), chained 16x16x4 (N cols 0..3 valid)
      v8f oacc = {0.f, 0.f, 0.f, 0.f, 0.f, 0.f, 0.f, 0.f};
      const int vt = t0 + (m & 3);            // clamped token column
      const float bm = (m < 4) ? 1.f : 0.f;   // zero the 12 unused N columns
#pragma unroll
      for (int u = 0; u < 4; ++u) {
        // A frag = P rows: P[cq=m][ck0], ck0 = 4u + 2*half. P^T lives in D
        // layout; transpose via lane permute (ds_bpermute) + half-select.
        const int src = m + ((u >= 2) ? 16 : 0);
        const int rA  = (4 * u) & 7;
        const float c0a = __shfl(st[rA],     src, 32);
        const float c1a = __shfl(st[rA + 2], src, 32);
        const float c0b = __shfl(st[rA + 1], src, 32);
        const float c1b = __shfl(st[rA + 3], src, 32);
        v2f pa;
        pa.x = half ? c1a : c0a;
        pa.y = half ? c1b : c0b;
        const int ck0 = 4 * u + 2 * half;
        v2f vb;
        vb.x = qkv[(128 +  ck0      * 4 + hd) * 16 + vt] * bm;
        vb.y = qkv[(128 + (ck0 + 1) * 4 + hd) * 16 + vt] * bm;
        oacc = __builtin_amdgcn_wmma_f32_16x16x4_f32(
            false, pa, false, vb, (short)0, oacc, false, false);
      }

      // O in D layout: lanes with column n<4 hold valid tokens
      if (m < 4) {
        const int tl  = m;
        const int po2 = (tl >> 1) * 8 + win * 2 + (tl & 1);
#pragma unroll
        for (int r = 0; r < 8; ++r) {
          const int ch = (r + 8 * half) * 4 + hd;   // out ch = c*HEADS + head
          obuf[ch * 16 + po2] = oacc[r];
        }
      }
    }
  }
  asm volatile("s_wait_dscnt 0x0" ::: "memory");

  // ---- Phase 3: async DMA output tile LDS -> global ----
#pragma unroll
  for (int it = 0; it < 8; ++it) {
    const int idx = it * 32 + lane;
    const int ch = idx >> 2, rr = (idx >> 1) & 1, qq = idx & 1;
    float* gp = out + imgBase + (size_t)ch * 65536
                + (y0 + rr) * 256 + x0 + qq * 4;
    const unsigned ldso = (unsigned)(uintptr_t)(obuf + ch * 16 + rr * 8 + qq * 4);
    asm volatile("global_store_async_from_lds_b128 %0, %1, off"
                 :: "v"(gp), "v"(ldso) : "memory");
  }
  asm volatile("s_wait_asynccnt 0x0" ::: "memory");
}

extern "C" void kernel_launch(void* const* d_in, const int* in_sizes, int n_in,
                              void* d_out, int out_size, void* d_ws, size_t ws_size,
                              hipStream_t stream) {
  (void)in_sizes; (void)n_in; (void)out_size; (void)d_ws; (void)ws_size;
  const float* x = (const float*)d_in[0];
  const float* w = (const float*)d_in[1];
  float* out = (float*)d_out;
  // 32768 tiles (8 images * 128 window-rows * 32 tile-cols), 2 waves per WG
  dim3 grid(32768 / WAVES_PER_WG), block(WAVES_PER_WG * 32);
  ca_fused_wmma<<<grid, block, 0, stream>>>(x, w, out);
}